// BDC_23794118820458
// MI455X (gfx1250) — compile-verified
//
#include <hip/hip_runtime.h>
#include <hip/hip_bf16.h>
#include <math.h>

// CDNA5 / gfx1250: wave32, fp32 WMMA path (V_WMMA_F32_16X16X4_F32).
typedef __attribute__((ext_vector_type(2))) float v2f;
typedef __attribute__((ext_vector_type(8))) float v8f;

#define DIMN       640
#define MLEN       100
#define LDS_STRIDE 102   // even -> 8B-aligned rows for ds_load_b64 fragments
#define NBLK2      20    // 640 / 32 super-tiles per side
#define NSUPER     (NBLK2 * NBLK2)            // 400 full super-tiles
#define NSUPER_UT  (NBLK2 * (NBLK2 + 1) / 2)  // 210 upper-triangular super-tiles
#define NTHREADS   1024
#define NWAVES     32

__device__ __forceinline__ v8f wmma_f32(v2f a, v2f b, v8f c) {
    // (neg_a, A, neg_b, B, c_mod, C, reuse_a, reuse_b)
    return __builtin_amdgcn_wmma_f32_16x16x4_f32(false, a, false, b,
                                                 (short)0, c, false, false);
}

__global__ __launch_bounds__(NTHREADS)
void bdc_wmma_kernel(const float* __restrict__ x,
                     const float* __restrict__ t,
                     float* __restrict__ out) {
    extern __shared__ float smem[];
    float* Xs   = smem;                       // 640 * 102 staged X[b]
    float* diag = Xs + DIMN * LDS_STRIDE;     // 640  diag(G)
    float* S    = diag + DIMN;                // 640  row==col sums of dcov
    float* tots = S + DIMN;                   // 1    total sum

    const int b   = blockIdx.x;
    const int tid = threadIdx.x;
    const float* __restrict__ xb = x + (size_t)b * (DIMN * MLEN);
    float* __restrict__ outb     = out + (size_t)b * (DIMN * DIMN);

    // ---- init sums, stage X[b] (256 KB) into 320KB WGP LDS ----
    for (int i = tid; i < DIMN; i += NTHREADS) S[i] = 0.f;
    if (tid == 0) tots[0] = 0.f;
    for (int idx = tid; idx < DIMN * MLEN; idx += NTHREADS) {
        int r = idx / MLEN;
        int c = idx - r * MLEN;
        Xs[r * LDS_STRIDE + c] = xb[idx];
    }
    __syncthreads();

    // ---- diag(G): 640 dot products of length 100, from LDS ----
    for (int r = tid; r < DIMN; r += NTHREADS) {
        const float* row = &Xs[r * LDS_STRIDE];
        float s = 0.f;
        for (int k = 0; k < MLEN; ++k) s = fmaf(row[k], row[k], s);
        diag[r] = s;
    }
    __syncthreads();

    const float et = expf(t[0]);

    const int wave = tid >> 5;
    const int lane = tid & 31;
    const int lg   = lane >> 4;   // half-wave: selects K pair (A/B) and M+8 (C/D)
    const int lr   = lane & 15;   // fragment row / output column

    // dcov element from Gram value g at (row r, col given by lr).
    // Native v_sqrt_f32 (1 ulp) instead of the correctly-rounded expansion.
    auto dcov_elem = [&](float g, int r, float dj) -> float {
        float dc = diag[r] + dj - 2.0f * g;
        dc = fmaxf(dc, 1e-4f);
        return __builtin_amdgcn_sqrtf(fmaf(et, dc, 1e-5f));
    };

    // =============== PHASE 1: symmetric sum pass (upper triangle only) =====
    // dcov is symmetric, so row sums == col sums; one array S[] gets
    // row-partials from each tile and (off-diagonal) col-partials for the
    // mirrored entries. No global memory traffic in this phase.
    for (int u = wave; u < NSUPER_UT; u += NWAVES) {
        int bi = 0, rem = u;                   // map u -> (bi <= bj)
        while (rem >= NBLK2 - bi) { rem -= NBLK2 - bi; ++bi; }
        const int bj = bi + rem;
        const int ti = bi * 32, tj = bj * 32;
        const bool diagblk = (bi == bj);

        v8f acc00 = {}, acc01 = {}, acc10 = {}, acc11 = {};
        const float* A0 = &Xs[(ti + lr)      * LDS_STRIDE + 2 * lg];
        const float* A1 = &Xs[(ti + 16 + lr) * LDS_STRIDE + 2 * lg];
        const float* B0 = &Xs[(tj + lr)      * LDS_STRIDE + 2 * lg];
        const float* B1 = &Xs[(tj + 16 + lr) * LDS_STRIDE + 2 * lg];
        #pragma unroll 5
        for (int k = 0; k < MLEN; k += 4) {
            v2f a0 = *(const v2f*)(A0 + k);
            v2f a1 = *(const v2f*)(A1 + k);
            v2f b0 = *(const v2f*)(B0 + k);
            v2f b1 = *(const v2f*)(B1 + k);
            acc00 = wmma_f32(a0, b0, acc00);
            acc01 = wmma_f32(a0, b1, acc01);
            acc10 = wmma_f32(a1, b0, acc10);
            acc11 = wmma_f32(a1, b1, acc11);
        }

        auto sum_tile = [&](v8f c, int ti2, int tj2) {
            const float dj = diag[tj2 + lr];
            float colpart = 0.f;
            #pragma unroll
            for (int v = 0; v < 8; ++v) {
                const int r = ti2 + v + 8 * lg;      // C/D: M = v + 8*lg
                const float dc = dcov_elem(c[v], r, dj);
                colpart += dc;
                float rp = dc;                        // 16-lane row reduce
                rp += __shfl_xor(rp, 1, 32);
                rp += __shfl_xor(rp, 2, 32);
                rp += __shfl_xor(rp, 4, 32);
                rp += __shfl_xor(rp, 8, 32);
                if (lr == 0) atomicAdd(&S[r], rp);    // ds_add_f32
            }
            if (!diagblk) {                           // mirrored entries' rows
                colpart += __shfl_xor(colpart, 16, 32);
                if (lg == 0) atomicAdd(&S[tj2 + lr], colpart);
            }
        };
        sum_tile(acc00, ti,      tj);
        sum_tile(acc01, ti,      tj + 16);
        sum_tile(acc10, ti + 16, tj);
        sum_tile(acc11, ti + 16, tj + 16);
    }

    __syncthreads();
    for (int i = tid; i < DIMN; i += NTHREADS) atomicAdd(tots, S[i]);
    __syncthreads();

    const float invd = 1.0f / (float)DIMN;
    const float tm   = tots[0] * (1.0f / ((float)DIMN * (float)DIMN));

    // =============== PHASE 2: recompute + fused centering + single store ===
    for (int blk = wave; blk < NSUPER; blk += NWAVES) {
        const int ti = (blk / NBLK2) * 32;
        const int tj = (blk % NBLK2) * 32;

        v8f acc00 = {}, acc01 = {}, acc10 = {}, acc11 = {};
        const float* A0 = &Xs[(ti + lr)      * LDS_STRIDE + 2 * lg];
        const float* A1 = &Xs[(ti + 16 + lr) * LDS_STRIDE + 2 * lg];
        const float* B0 = &Xs[(tj + lr)      * LDS_STRIDE + 2 * lg];
        const float* B1 = &Xs[(tj + 16 + lr) * LDS_STRIDE + 2 * lg];
        #pragma unroll 5
        for (int k = 0; k < MLEN; k += 4) {
            v2f a0 = *(const v2f*)(A0 + k);
            v2f a1 = *(const v2f*)(A1 + k);
            v2f b0 = *(const v2f*)(B0 + k);
            v2f b1 = *(const v2f*)(B1 + k);
            acc00 = wmma_f32(a0, b0, acc00);
            acc01 = wmma_f32(a0, b1, acc01);
            acc10 = wmma_f32(a1, b0, acc10);
            acc11 = wmma_f32(a1, b1, acc11);
        }

        auto store_tile = [&](v8f c, int ti2, int tj2) {
            const int col   = tj2 + lr;
            const float dj  = diag[col];
            const float cmm = S[col] * invd - tm;     // col mean - total mean
            #pragma unroll
            for (int v = 0; v < 8; ++v) {
                const int r = ti2 + v + 8 * lg;
                const float dc = dcov_elem(c[v], r, dj);
                // streaming result, never re-read: non-temporal store (TH=NT)
                __builtin_nontemporal_store(dc - S[r] * invd - cmm,
                                            &outb[r * DIMN + col]);
            }
        };
        store_tile(acc00, ti,      tj);
        store_tile(acc01, ti,      tj + 16);
        store_tile(acc10, ti + 16, tj);
        store_tile(acc11, ti + 16, tj + 16);
    }
}

extern "C" void kernel_launch(void* const* d_in, const int* in_sizes, int n_in,
                              void* d_out, int out_size, void* d_ws, size_t ws_size,
                              hipStream_t stream) {
    (void)n_in; (void)out_size; (void)d_ws; (void)ws_size;
    const float* x = (const float*)d_in[0];
    const float* t = (const float*)d_in[1];
    float* out = (float*)d_out;

    const int batches = in_sizes[0] / (DIMN * MLEN);   // 128 for the reference
    const size_t shmem = (size_t)(DIMN * LDS_STRIDE + 2 * DIMN + 1) * sizeof(float);

    bdc_wmma_kernel<<<dim3(batches), dim3(NTHREADS), shmem, stream>>>(x, t, out);
}